// Featurizer_40054865002568
// MI455X (gfx1250) — compile-verified
//
#include <hip/hip_runtime.h>

typedef __attribute__((ext_vector_type(2))) float v2f;
typedef __attribute__((ext_vector_type(8))) float v8f;

#define RC_F    6.0f
#define PI_F    3.14159265358979f
#define MAXN    352   // padded neighbor-array capacity (>= ceil(319/16)*16)
#define MAXW    12    // max waves per block (352/32)

__device__ __forceinline__ float fc_cut(float r) {
    // fc(r) = 0.5*(cos(pi*r/Rc)+1), callers handle r >= Rc
    return 0.5f * (__cosf(PI_F * (1.0f / RC_F) * r) + 1.0f);
}

__global__ __launch_bounds__(MAXN) void acsf_kernel(const float* __restrict__ zin,
                                                    const float* __restrict__ xyz,
                                                    float* __restrict__ out,
                                                    int N)
{
    // Fragment planes: plane 0 = (dx,dy) for lanes 0-15 (K=0,1),
    //                  plane 1 = (dz, 0) for lanes 16-31 (K=2,3).
    __shared__ v2f    sAB[2][MAXN];
    __shared__ float4 sP[MAXN];          // (r, r2, fc, 0) -> one ds_load_b128
    __shared__ int    wcnt[MAXW], woff[MAXW];
    __shared__ int    sM;
    __shared__ float  red[MAXW][21];

    const int i    = blockIdx.x;
    const int tid  = threadIdx.x;
    const int wid  = tid >> 5;
    const int lane = tid & 31;
    const int nw   = blockDim.x >> 5;

    const float xi = xyz[3 * i + 0];
    const float yi = xyz[3 * i + 1];
    const float zi = xyz[3 * i + 2];

    // ---- init padded neighbor slots (padding contributes exactly zero) ----
    for (int p = tid; p < MAXN; p += blockDim.x) {
        sAB[0][p] = (v2f){0.0f, 0.0f};
        sAB[1][p] = (v2f){0.0f, 0.0f};
        sP[p]     = make_float4(1.0f, 0.0f, 0.0f, 0.0f);   // r=1 avoids rcp(0)
    }

    float acc[21];
    #pragma unroll
    for (int c = 0; c < 21; ++c) acc[c] = 0.0f;

    // ---- phase 1: pair quantities, radial features, neighbor flag ----
    float dx = 0.f, dy = 0.f, dz = 0.f, r = 1.f, r2 = 0.f, fc = 0.f;
    bool nbr = false;
    if (tid < N && tid != i) {
        dx = xyz[3 * tid + 0] - xi;
        dy = xyz[3 * tid + 1] - yi;
        dz = xyz[3 * tid + 2] - zi;
        r2 = dx * dx + dy * dy + dz * dz;
        r  = __builtin_amdgcn_sqrtf(r2);
        if (r < RC_F) {
            nbr = true;
            fc  = fc_cut(r);
            acc[0] = fc;                                  // G1
            const float g2eta[8] = {0.5f, 1.0f, 2.0f, 4.0f, 0.5f, 1.0f, 2.0f, 4.0f};
            const float g2rs [8] = {0.0f, 0.0f, 0.0f, 0.0f, 3.0f, 3.0f, 3.0f, 3.0f};
            #pragma unroll
            for (int e = 0; e < 8; ++e) {                 // G2
                float d = r - g2rs[e];
                acc[1 + e] = __expf(-g2eta[e] * d * d) * fc;
            }
            const float g3k[4] = {0.5f, 1.0f, 1.5f, 2.0f};
            #pragma unroll
            for (int m = 0; m < 4; ++m)                   // G3
                acc[9 + m] = __cosf(g3k[m] * r) * fc;
        }
    }

    // ---- deterministic neighbor compaction: ballot + prefix scan ----
    unsigned m32 = (unsigned)__ballot(nbr);               // wave32: low 32 bits
    if (lane == 0) wcnt[wid] = __popc(m32);
    __syncthreads();
    if (tid == 0) {
        int s = 0;
        for (int w = 0; w < nw; ++w) { woff[w] = s; s += wcnt[w]; }
        sM = s;
    }
    __syncthreads();
    if (nbr) {
        int idx = woff[wid] + __popc(m32 & ((1u << lane) - 1u));
        sAB[0][idx] = (v2f){dx, dy};
        sAB[1][idx] = (v2f){dz, 0.0f};
        sP[idx]     = make_float4(r, r2, fc, 0.0f);
    }
    __syncthreads();

    const int M      = sM;
    const int Mp     = (M + 15) & ~15;
    const int ntj    = Mp >> 4;
    const int ntiles = ntj * ntj;

    // ---- phase 2: per-atom neighbor Gram matrix via WMMA f32 16x16x4 ----
    // S = D * D^T, D = [dx dy dz 0] (Mp x 4). cos = S/(r_j r_k),
    // r2_jk = r2_j + r2_k - 2S. Tiles round-robin across waves (wave-uniform
    // loop => EXEC all-ones at the WMMA; inner math is branchless).
    const int  l16   = lane & 15;
    const bool hi    = lane >= 16;
    const v2f* plane = sAB[hi ? 1 : 0];   // address select, not value select
    for (int t = wid; t < ntiles; t += nw) {
        const int ta = t / ntj;
        const int tb = t - ta * ntj;
        const int jj = ta * 16 + l16;
        const int kk = tb * 16 + l16;

        v2f a = plane[jj];                // single ds_load_b64 each
        v2f b = plane[kk];
        v8f cfrag = {};
        cfrag = __builtin_amdgcn_wmma_f32_16x16x4_f32(
            false, a, false, b, (short)0, cfrag, false, false);

        const float4 Pk  = sP[kk];        // ds_load_b128
        const float  rk  = Pk.x, r2k = Pk.y, fck = Pk.z;
        if (fck > 0.0f) {                 // skip fully-padded / far columns
            const int rowBase = ta * 16 + (hi ? 8 : 0);
            #pragma unroll
            for (int v = 0; v < 8; ++v) {
                const int    row = rowBase + v;
                const float4 Pj  = sP[row];               // ds_load_b128
                const float  rj  = Pj.x, r2j = Pj.y, fcj = Pj.z;
                const float  dot = cfrag[v];
                const float cosv = dot * __builtin_amdgcn_rcpf(rj * rk);
                const float r2p  = r2j + r2k;                     // rij^2+rik^2
                const float r2jk = fmaxf(r2p - 2.0f * dot, 0.0f); // rjk^2
                const float rjk  = __builtin_amdgcn_sqrtf(r2jk);
                float fcjk = fc_cut(rjk);
                fcjk = (rjk < RC_F) ? fcjk : 0.0f;

                // exp(-0.5x) = exp(-0.1x)^5 -> only 2 transcendental exps
                const float e1p = __expf(-0.1f * r2p);
                const float e1j = __expf(-0.1f * r2jk);
                float q = e1p * e1p;  const float e5p = q * q * e1p;
                q = e1j * e1j;        const float e5j = q * q * e1j;
                const float e1t = e1p * e1j;
                const float e5t = e5p * e5j;

                // padded rows have fcj==0 -> zero contribution; only j==k
                // needs an explicit (branchless) mask.
                float fc2v = fcj * fck;
                fc2v = (row == kk) ? 0.0f : fc2v;
                const float fc3v = fc2v * fcjk;
                const float op = 1.0f + cosv, om = 1.0f - cosv;
                const float p2 = om * om;
                const float o2 = op * op;
                const float p3 = o2 * o2;   // (1+c)^4
                const float p4 = p2 * p2;   // (1-c)^4

                acc[13] += op * e1t * fc3v;   // G4 (0.1,1,+1)
                acc[14] += p2 * e1t * fc3v;   // G4 (0.1,2,-1)
                acc[15] += p3 * e5t * fc3v;   // G4 (0.5,4,+1)
                acc[16] += p4 * e5t * fc3v;   // G4 (0.5,4,-1)
                acc[17] += op * e1p * fc2v;   // G5 (0.1,1,+1)
                acc[18] += p2 * e1p * fc2v;   // G5 (0.1,2,-1)
                acc[19] += p3 * e5p * fc2v;   // G5 (0.5,4,+1)
                acc[20] += p4 * e5p * fc2v;   // G5 (0.5,4,-1)
            }
        }
    }

    // ---- phase 3: wave32 shfl reduction + cross-wave LDS reduction ----
    #pragma unroll
    for (int c = 0; c < 21; ++c) {
        float v = acc[c];
        #pragma unroll
        for (int o = 16; o > 0; o >>= 1) v += __shfl_xor(v, o, 32);
        if (lane == 0) red[wid][c] = v;
    }
    __syncthreads();
    if (tid < 21) {
        float s = 0.0f;
        for (int w = 0; w < nw; ++w) s += red[w][tid];
        float sc = 1.0f;                        // 2^(1-zeta)*0.5 for angular cols
        if (tid >= 13) {
            int q = (tid - 13) & 3;             // zeta = 1,2,4,4
            sc = (q == 0) ? 0.5f : ((q == 1) ? 0.25f : 0.0625f);
        }
        out[i * 22 + 1 + tid] = s * sc;
    }
    if (tid == 0) out[i * 22 + 0] = zin[i];
}

extern "C" void kernel_launch(void* const* d_in, const int* in_sizes, int n_in,
                              void* d_out, int out_size, void* d_ws, size_t ws_size,
                              hipStream_t stream) {
    const float* z   = (const float*)d_in[0];   // atomic_numbers [N,1] f32
    const float* xyz = (const float*)d_in[1];   // coordinates   [N,3] f32
    float* out = (float*)d_out;                 // [N,22] f32
    int N = in_sizes[0];                        // 320
    int block = ((N + 31) / 32) * 32;           // multiple of 32 (wave32)
    if (block > MAXN) block = MAXN;
    acsf_kernel<<<N, block, 0, stream>>>(z, xyz, out, N);
}